// GAT_CS_83631603187928
// MI455X (gfx1250) — compile-verified
//
#include <hip/hip_runtime.h>
#include <math.h>

typedef float v2f __attribute__((ext_vector_type(2)));
typedef float v8f __attribute__((ext_vector_type(8)));

#define B_ 16
#define N_ 2048
#define D_ 20
#define H_ 128
#define LEAKY 0.01f

#define JC 32            // j rows per LDS chunk
#define RS 144           // LDS row stride in floats (bank-conflict-free padding)

__device__ __forceinline__ float lrelu(float x) { return x >= 0.f ? x : LEAKY * x; }

__device__ __forceinline__ v8f wmma_f32(v2f a, v2f b, v8f c) {
    // D = A(16x4 f32) * B(4x16 f32) + C(16x16 f32)
    return __builtin_amdgcn_wmma_f32_16x16x4_f32(false, a, false, b, (short)0, c,
                                                 false, false);
}

#if __has_builtin(__builtin_amdgcn_s_wait_asynccnt)
#define WAIT_ASYNC0() __builtin_amdgcn_s_wait_asynccnt(0)
#else
#define WAIT_ASYNC0() asm volatile("s_wait_asynccnt 0x0" ::: "memory")
#endif

// Async copy of 16 bytes: global -> LDS (per active lane). ASYNCcnt-tracked.
__device__ __forceinline__ void async_g2l_b128(unsigned lds_byte_addr,
                                               const void* gaddr) {
    asm volatile("global_load_async_to_lds_b128 %0, %1, off"
                 :: "v"(lds_byte_addr),
                    "v"((unsigned long long)(uintptr_t)gaddr)
                 : "memory");
}

// ---------------------------------------------------------------------------
// hidden0 = x @ W_in + b_in        (M = B*N rows, K = D = 20, N = H = 128)
// One wave per 16-row tile; 8 column tiles; K = 5 steps of 4.
// ---------------------------------------------------------------------------
__global__ void k_gemm_in(const float* __restrict__ x,
                          const float* __restrict__ W_in,
                          const float* __restrict__ b_in,
                          float* __restrict__ hidden) {
    const int lane = threadIdx.x;            // 0..31
    const int n16  = lane & 15;
    const int half = lane >> 4;
    const int m0   = blockIdx.x * 16;        // row tile base in [0, B*N)

    // A fragments: lane holds row m=n16, K = k5*4 + half*2 + {0,1}
    v2f afrag[5];
#pragma unroll
    for (int k5 = 0; k5 < 5; ++k5) {
        const int k = k5 * 4 + half * 2;
        const float* xr = x + (size_t)(m0 + n16) * D_ + k;
        afrag[k5].x = xr[0];
        afrag[k5].y = xr[1];
    }

#pragma unroll
    for (int nt = 0; nt < 8; ++nt) {
        v8f acc = {};
#pragma unroll
        for (int k5 = 0; k5 < 5; ++k5) {
            const int k = k5 * 4 + half * 2;
            v2f bfrag;
            bfrag.x = W_in[(size_t)(k + 0) * H_ + nt * 16 + n16];
            bfrag.y = W_in[(size_t)(k + 1) * H_ + nt * 16 + n16];
            acc = wmma_f32(afrag[k5], bfrag, acc);
        }
        const float bias = b_in[nt * 16 + n16];
#pragma unroll
        for (int v = 0; v < 8; ++v) {
            const int m = v + 8 * half;
            hidden[(size_t)(m0 + m) * H_ + nt * 16 + n16] = acc[v] + bias;
        }
    }
}

// ---------------------------------------------------------------------------
// Fold W_t/a/b_t:  v_j = W_t @ a[:H], v_i = W_t @ a[H:], c = b_t . a
// vv layout: [0..127]=v_j  [128..255]=v_i  [256]=c_j  [257]=c_i
// ---------------------------------------------------------------------------
__global__ void k_vec(const float* __restrict__ W_t,
                      const float* __restrict__ b_t,
                      const float* __restrict__ a,
                      float* __restrict__ vv) {
    const int h = threadIdx.x;   // 128 threads
    float vj = 0.f, vi = 0.f;
    for (int k = 0; k < H_; ++k) {
        const float w = W_t[(size_t)h * H_ + k];
        vj += w * a[k];
        vi += w * a[H_ + k];
    }
    vv[h]       = vj;
    vv[128 + h] = vi;
    if (h == 0) { float c = 0.f; for (int k = 0; k < H_; ++k) c += b_t[k] * a[k];       vv[256] = c; }
    if (h == 1) { float c = 0.f; for (int k = 0; k < H_; ++k) c += b_t[k] * a[H_ + k];  vv[257] = c; }
}

// ---------------------------------------------------------------------------
// s_j[r] = hidden[r,:] . v_j + c_j ;  s_i[r] = hidden[r,:] . v_i + c_i
// ---------------------------------------------------------------------------
__global__ void k_scores(const float* __restrict__ hidden,
                         const float* __restrict__ vv,
                         float* __restrict__ s_i,
                         float* __restrict__ s_j) {
    const int r = blockIdx.x * blockDim.x + threadIdx.x;   // row in [0, B*N)
    const float* hr = hidden + (size_t)r * H_;
    float aj = vv[256], ai = vv[257];
    for (int h = 0; h < H_; ++h) {
        const float hv = hr[h];
        aj += hv * vv[h];
        ai += hv * vv[128 + h];
    }
    s_j[r] = aj;
    s_i[r] = ai;
}

// ---------------------------------------------------------------------------
// Per-batch max over s_j (exact softmax row max = lrelu(s_i + max_j s_j)).
// ---------------------------------------------------------------------------
__global__ void k_maxsj(const float* __restrict__ s_j, float* __restrict__ maxsj) {
    __shared__ float red[256];
    const int b = blockIdx.x;
    float m = -INFINITY;
    for (int n = threadIdx.x; n < N_; n += 256) m = fmaxf(m, s_j[(size_t)b * N_ + n]);
    red[threadIdx.x] = m;
    __syncthreads();
    for (int s = 128; s > 0; s >>= 1) {
        if (threadIdx.x < s) red[threadIdx.x] = fmaxf(red[threadIdx.x], red[threadIdx.x + s]);
        __syncthreads();
    }
    if (threadIdx.x == 0) maxsj[b] = red[0];
}

// ---------------------------------------------------------------------------
// out = softmax(lrelu(s_i + s_j)) @ hidden + hidden
// 4 waves/block; each wave one 16-row i-tile with a 16x128 fp32 accumulator.
// V tiles staged in LDS via async global->LDS copies, double buffered; all
// four waves share each 32x128 V tile. P tiles built in-register (rank-1).
// ---------------------------------------------------------------------------
__global__ void __launch_bounds__(128) k_attn(const float* __restrict__ hidden,
                                              const float* __restrict__ s_i,
                                              const float* __restrict__ s_j,
                                              const float* __restrict__ maxsj,
                                              float* __restrict__ out) {
    __shared__ float smem[2 * JC * RS];   // 2 x 18KB double buffer

    const int tid  = threadIdx.x;
    const int lane = tid & 31;
    const int wave = tid >> 5;
    const int n16  = lane & 15;
    const int half = lane >> 4;

    const int blocksPerB = N_ / 64;                          // 32
    const int b  = blockIdx.x / blocksPerB;
    const int i0 = (blockIdx.x % blocksPerB) * 64 + wave * 16;

    const float* hB  = hidden + (size_t)b * N_ * H_;
    const float* sjB = s_j + (size_t)b * N_;

    const float si = s_i[(size_t)b * N_ + i0 + n16];         // A-frag row m = n16
    const float Mi = lrelu(si + maxsj[b]);                    // exact row max
    float rowsum = 0.f;

    v8f acc[8];
#pragma unroll
    for (int nt = 0; nt < 8; ++nt) acc[nt] = (v8f){};

    // Issue async fill of one 32x128 chunk into LDS buffer `buf`.
    // 128 threads x 16B = 4 rows per round; 8 rounds cover 32 rows.
    auto issue_fill = [&](int j0, int buf) {
#pragma unroll
        for (int rnd = 0; rnd < 8; ++rnd) {
            const int row = rnd * 4 + wave;
            const unsigned lds =
                (unsigned)(uintptr_t)&smem[buf * JC * RS + row * RS + lane * 4];
            const float* g = hB + (size_t)(j0 + row) * H_ + lane * 4;
            async_g2l_b128(lds, g);
        }
    };

    // Consume one chunk from LDS buffer `buf`.
    auto compute = [&](int j0, int buf) {
        const float* bufp = &smem[buf * JC * RS];
        for (int ks = 0; ks < JC / 4; ++ks) {
            const int rl = ks * 4 + half * 2;                // local K rows
            const int kb = j0 + rl;                          // global j
            v2f p;
            p.x = __expf(lrelu(si + sjB[kb])     - Mi);
            p.y = __expf(lrelu(si + sjB[kb + 1]) - Mi);
            rowsum += p.x + p.y;

            const float* r0 = bufp + rl * RS + n16;          // row kb, col n16
#pragma unroll
            for (int nt = 0; nt < 8; ++nt) {
                v2f bf;
                bf.x = r0[nt * 16];                          // row kb
                bf.y = r0[nt * 16 + RS];                     // row kb+1
                acc[nt] = wmma_f32(p, bf, acc[nt]);
            }
        }
    };

    issue_fill(0, 0);
    WAIT_ASYNC0();
    __syncthreads();

    const int nch = N_ / JC;                                 // 64
    for (int c = 0; c < nch; ++c) {
        if (c + 1 < nch) issue_fill((c + 1) * JC, (c + 1) & 1);
        compute(c * JC, c & 1);
        WAIT_ASYNC0();            // own async copies done
        __syncthreads();          // publish LDS writes to all waves
    }

    // combine the two half-lane partial row sums (rows live at lanes m and m+16)
    rowsum += __shfl_xor(rowsum, 16, 32);

    float inv[8];
#pragma unroll
    for (int v = 0; v < 8; ++v) {
        const int m = v + 8 * half;                          // acc row for this lane
        inv[v] = 1.0f / __shfl(rowsum, m, 32);
    }

#pragma unroll
    for (int nt = 0; nt < 8; ++nt) {
#pragma unroll
        for (int v = 0; v < 8; ++v) {
            const int m = v + 8 * half;
            const size_t idx = (size_t)(i0 + m) * H_ + nt * 16 + n16;
            out[(size_t)b * N_ * H_ + idx] = acc[nt][v] * inv[v] + hB[idx];
        }
    }
}

// ---------------------------------------------------------------------------
extern "C" void kernel_launch(void* const* d_in, const int* in_sizes, int n_in,
                              void* d_out, int out_size, void* d_ws, size_t ws_size,
                              hipStream_t stream) {
    const float* x    = (const float*)d_in[0];
    const float* W_in = (const float*)d_in[1];
    const float* b_in = (const float*)d_in[2];
    const float* W_t  = (const float*)d_in[3];
    const float* b_t  = (const float*)d_in[4];
    const float* a    = (const float*)d_in[5];
    float* outp = (float*)d_out;

    const size_t BNH = (size_t)B_ * N_ * H_;                 // 4,194,304 floats
    const size_t BN  = (size_t)B_ * N_;

    float* ws    = (float*)d_ws;
    float* h0    = ws;                 // layer-0 hidden
    float* h1    = h0 + BNH;           // layer-1 hidden
    float* s_i   = h1 + BNH;
    float* s_j   = s_i + BN;
    float* vv    = s_j + BN;           // 258 floats (padded to 272)
    float* maxsj = vv + 272;           // B_ floats

    // hidden0 = x @ W_in + b_in
    k_gemm_in<<<(B_ * N_) / 16, 32, 0, stream>>>(x, W_in, b_in, h0);

    // fold W_t, b_t, a into per-row score vectors (layer-invariant)
    k_vec<<<1, 128, 0, stream>>>(W_t, b_t, a, vv);

    // ---- layer 1 ----
    k_scores<<<(B_ * N_) / 256, 256, 0, stream>>>(h0, vv, s_i, s_j);
    k_maxsj<<<B_, 256, 0, stream>>>(s_j, maxsj);
    k_attn<<<B_ * (N_ / 64), 128, 0, stream>>>(h0, s_i, s_j, maxsj, h1);

    // ---- layer 2 ----
    k_scores<<<(B_ * N_) / 256, 256, 0, stream>>>(h1, vv, s_i, s_j);
    k_maxsj<<<B_, 256, 0, stream>>>(s_j, maxsj);
    k_attn<<<B_ * (N_ / 64), 128, 0, stream>>>(h1, s_i, s_j, maxsj, outp);
}